// TinyTransformer_23940147708300
// MI455X (gfx1250) — compile-verified
//
#include <hip/hip_runtime.h>
#include <hip/hip_bf16.h>
#include <math.h>

typedef __attribute__((ext_vector_type(2))) float v2f;
typedef __attribute__((ext_vector_type(4))) float v4f;
typedef __attribute__((ext_vector_type(8))) float v8f;

#define WAVES_PER_BLK 4
#define BATCH_PER_WAVE 8
#define ROWS 48           // 8 batches * 6 seq positions
#define SEQ 6
#define HID 8
#define FFD 16
#define LDS_PER_WAVE 2304 // floats: x(384)+q(384)+k(384)+v(384)+a(768)

// D = A(16x4 f32) * B(4x16 f32) + C(16x16 f32), full fp32 WMMA
__device__ __forceinline__ v8f wmma4(v2f a, v2f b, v8f c) {
  return __builtin_amdgcn_wmma_f32_16x16x4_f32(
      /*neg_a=*/false, a, /*neg_b=*/false, b,
      /*c_mod=*/(short)0, c, /*reuse_a=*/false, /*reuse_b=*/false);
}

// A-matrix 16x4 fragment from row-major LDS buffer.
// ISA layout: lanes 0-15 hold row M=lane, VGPR0/1 = K0,K1; lanes 16-31 = K2,K3.
__device__ __forceinline__ v2f load_a_frag(const float* buf, int stride,
                                           int tile, int kb, int lane) {
  int row = tile * 16 + (lane & 15);
  int col = kb * 4 + ((lane >> 4) << 1);
  const float* p = buf + row * stride + col;
  v2f a; a.x = p[0]; a.y = p[1];
  return a;
}

// B-matrix 4x16 fragment for x @ W^T : B[k][n] = W[n][k]; cols n>=N zeroed.
__device__ __forceinline__ v2f load_b_frag(const float* __restrict__ W, int N,
                                           int K, int kb, int lane) {
  int n = lane & 15;
  int col = kb * 4 + ((lane >> 4) << 1);
  v2f b;
  if (n < N) { b.x = W[n * K + col]; b.y = W[n * K + col + 1]; }
  else       { b.x = 0.f;            b.y = 0.f; }
  return b;
}

// C/D 16x16 fragment -> row-major LDS. VGPR j: lanes0-15 M=j, lanes16-31 M=j+8.
__device__ __forceinline__ void store_c_frag(float* buf, int stride, int ncols,
                                             int tile, v8f c, int lane) {
  int n = lane & 15;
  int m0 = tile * 16 + ((lane >> 4) << 3);
  if (n < ncols) {
#pragma unroll
    for (int j = 0; j < 8; ++j) buf[(m0 + j) * stride + n] = c[j];
  }
}

__global__ __launch_bounds__(128) void tiny_transformer_fused(
    const int* __restrict__ ids, const float* __restrict__ tok_emb,
    const float* __restrict__ pos_emb,
    const float* __restrict__ Wq, const float* __restrict__ bq,
    const float* __restrict__ Wk, const float* __restrict__ bk,
    const float* __restrict__ Wv, const float* __restrict__ bv,
    const float* __restrict__ Wo, const float* __restrict__ bo,
    const float* __restrict__ W1, const float* __restrict__ b1,
    const float* __restrict__ W2, const float* __restrict__ b2,
    const float* __restrict__ g1, const float* __restrict__ be1,
    const float* __restrict__ g2, const float* __restrict__ be2,
    float* __restrict__ out, int Btot) {
  __shared__ float smem[WAVES_PER_BLK * LDS_PER_WAVE];

  const int lane = threadIdx.x & 31;
  const int wave = threadIdx.x >> 5;
  const int blk_b0 = blockIdx.x * (WAVES_PER_BLK * BATCH_PER_WAVE);
  // Whole-block guard (uniform): drop incomplete tail so EXEC stays all-1 for WMMA.
  if (blk_b0 + WAVES_PER_BLK * BATCH_PER_WAVE > Btot) return;
  const int b0 = blk_b0 + wave * BATCH_PER_WAVE;

  float* xbuf = smem + wave * LDS_PER_WAVE; // [48][8]  x, later h
  float* qbuf = xbuf + 384;                 // [48][8]
  float* kbuf = qbuf + 384;                 // [48][8]  k, later o / ffn2
  float* vbuf = kbuf + 384;                 // [48][8]
  float* abuf = vbuf + 384;                 // [48][16] attn / gelu

  // ---- Preload weight B-fragments + column-biases (weights are L2-resident) ----
  v2f fWq[2], fWk[2], fWv[2], fWo[2], fW1[2], fW2[4];
#pragma unroll
  for (int kb = 0; kb < 2; ++kb) {
    fWq[kb] = load_b_frag(Wq, 8, 8, kb, lane);
    fWk[kb] = load_b_frag(Wk, 8, 8, kb, lane);
    fWv[kb] = load_b_frag(Wv, 8, 8, kb, lane);
    fWo[kb] = load_b_frag(Wo, 8, 8, kb, lane);
    fW1[kb] = load_b_frag(W1, 16, 8, kb, lane);
  }
#pragma unroll
  for (int kb = 0; kb < 4; ++kb) fW2[kb] = load_b_frag(W2, 8, 16, kb, lane);

  const int ncol = lane & 15;
  const float cbq = (ncol < 8) ? bq[ncol] : 0.f;
  const float cbk = (ncol < 8) ? bk[ncol] : 0.f;
  const float cbv = (ncol < 8) ? bv[ncol] : 0.f;
  const float cbo = (ncol < 8) ? bo[ncol] : 0.f;
  const float cb1 = b1[ncol];
  const float cb2 = (ncol < 8) ? b2[ncol] : 0.f;

  // ---- P1: embedding gather  x = tok_emb[id] + pos_emb[s] ----
  for (int r = lane; r < ROWS; r += 32) {
    int bl = r / SEQ, s = r - bl * SEQ;
    int id = ids[(b0 + bl) * SEQ + s];
    const float* te = tok_emb + id * HID;
    const float* pe = pos_emb + s * HID;
    float* xr = xbuf + r * HID;
#pragma unroll
    for (int c = 0; c < HID; ++c) xr[c] = te[c] + pe[c];
  }
  __syncthreads();

  // ---- P2: q,k,v projections via WMMA (K=8 -> 2 chained k-steps) ----
#pragma unroll
  for (int t = 0; t < 3; ++t) {
    v8f cq = {}; v8f ck = {}; v8f cv = {};
#pragma unroll
    for (int kb = 0; kb < 2; ++kb) {
      v2f a = load_a_frag(xbuf, HID, t, kb, lane);
      cq = wmma4(a, fWq[kb], cq);
      ck = wmma4(a, fWk[kb], ck);
      cv = wmma4(a, fWv[kb], cv);
    }
#pragma unroll
    for (int j = 0; j < 8; ++j) { cq[j] += cbq; ck[j] += cbk; cv[j] += cbv; }
    store_c_frag(qbuf, HID, 8, t, cq, lane);
    store_c_frag(kbuf, HID, 8, t, ck, lane);
    store_c_frag(vbuf, HID, 8, t, cv, lane);
  }
  __syncthreads();

  // ---- P3: attention, 1 lane = 1 (b,h,s) query; 96 queries = 3 rounds ----
#pragma unroll
  for (int jq = 0; jq < 3; ++jq) {
    int qi = lane + 32 * jq;
    int bl = qi / 12;
    int rem = qi - bl * 12;
    int hh = rem / SEQ;
    int s = rem - hh * SEQ;
    const float* qr = qbuf + (bl * SEQ + s) * HID + hh * 4;
    float q0 = qr[0], q1 = qr[1], q2 = qr[2], q3 = qr[3];
    float sc[SEQ]; float mx = -1e30f;
#pragma unroll
    for (int tt = 0; tt < SEQ; ++tt) {
      const float* kr = kbuf + (bl * SEQ + tt) * HID + hh * 4;
      float d = (q0 * kr[0] + q1 * kr[1] + q2 * kr[2] + q3 * kr[3]) * 0.5f; // 1/sqrt(4)
      sc[tt] = d; mx = fmaxf(mx, d);
    }
    float sum = 0.f;
#pragma unroll
    for (int tt = 0; tt < SEQ; ++tt) { float e = __expf(sc[tt] - mx); sc[tt] = e; sum += e; }
    float inv = 1.f / sum;
    float a0 = 0.f, a1 = 0.f, a2 = 0.f, a3 = 0.f;
#pragma unroll
    for (int tt = 0; tt < SEQ; ++tt) {
      const float* vr = vbuf + (bl * SEQ + tt) * HID + hh * 4;
      float w = sc[tt] * inv;
      a0 += w * vr[0]; a1 += w * vr[1]; a2 += w * vr[2]; a3 += w * vr[3];
    }
    float* ar = abuf + (bl * SEQ + s) * FFD + hh * 4;
    ar[0] = a0; ar[1] = a1; ar[2] = a2; ar[3] = a3;
  }
  __syncthreads();

  // ---- P4: output projection o = attn @ Wo^T + bo  -> kbuf ----
#pragma unroll
  for (int t = 0; t < 3; ++t) {
    v8f co = {};
#pragma unroll
    for (int kb = 0; kb < 2; ++kb) {
      v2f a = load_a_frag(abuf, FFD, t, kb, lane);
      co = wmma4(a, fWo[kb], co);
    }
#pragma unroll
    for (int j = 0; j < 8; ++j) co[j] += cbo;
    store_c_frag(kbuf, HID, 8, t, co, lane);
  }
  __syncthreads();

  // ---- P5: h = LN1(x + o)  -> xbuf ----
  for (int r = lane; r < ROWS; r += 32) {
    const float* xr = xbuf + r * HID;
    const float* orow = kbuf + r * HID;
    float v[8]; float mu = 0.f;
#pragma unroll
    for (int c = 0; c < 8; ++c) { v[c] = xr[c] + orow[c]; mu += v[c]; }
    mu *= 0.125f;
    float var = 0.f;
#pragma unroll
    for (int c = 0; c < 8; ++c) { float d = v[c] - mu; var += d * d; }
    var *= 0.125f;
    float rs = rsqrtf(var + 1e-5f);
    float* hr = xbuf + r * HID;
#pragma unroll
    for (int c = 0; c < 8; ++c) hr[c] = (v[c] - mu) * rs * g1[c] + be1[c];
  }
  __syncthreads();

  // ---- P6: ffn1 = gelu(h @ W1^T + b1), N=16 fills the tile  -> abuf ----
#pragma unroll
  for (int t = 0; t < 3; ++t) {
    v8f c1 = {};
#pragma unroll
    for (int kb = 0; kb < 2; ++kb) {
      v2f a = load_a_frag(xbuf, HID, t, kb, lane);
      c1 = wmma4(a, fW1[kb], c1);
    }
#pragma unroll
    for (int j = 0; j < 8; ++j) {
      float u = c1[j] + cb1;
      c1[j] = 0.5f * u * (1.f + erff(u * 0.70710678118f)); // exact gelu
    }
    store_c_frag(abuf, FFD, 16, t, c1, lane);
  }
  __syncthreads();

  // ---- P7: ffn2 = gelu_out @ W2^T + b2 (K=16 -> 4 k-steps)  -> kbuf ----
#pragma unroll
  for (int t = 0; t < 3; ++t) {
    v8f c2 = {};
#pragma unroll
    for (int kb = 0; kb < 4; ++kb) {
      v2f a = load_a_frag(abuf, FFD, t, kb, lane);
      c2 = wmma4(a, fW2[kb], c2);
    }
#pragma unroll
    for (int j = 0; j < 8; ++j) c2[j] += cb2;
    store_c_frag(kbuf, HID, 8, t, c2, lane);
  }
  __syncthreads();

  // ---- P8: out = LN2(h + ffn2), coalesced float4 stores ----
  float* dst = out + (size_t)b0 * SEQ * HID;
  for (int r = lane; r < ROWS; r += 32) {
    const float* hr = xbuf + r * HID;
    const float* fr = kbuf + r * HID;
    float v[8]; float mu = 0.f;
#pragma unroll
    for (int c = 0; c < 8; ++c) { v[c] = hr[c] + fr[c]; mu += v[c]; }
    mu *= 0.125f;
    float var = 0.f;
#pragma unroll
    for (int c = 0; c < 8; ++c) { float d = v[c] - mu; var += d * d; }
    var *= 0.125f;
    float rs = rsqrtf(var + 1e-5f);
    v4f o0, o1;
#pragma unroll
    for (int c = 0; c < 4; ++c) {
      o0[c] = (v[c] - mu) * rs * g2[c] + be2[c];
      o1[c] = (v[c + 4] - mu) * rs * g2[c + 4] + be2[c + 4];
    }
    *(v4f*)(dst + r * HID) = o0;
    *(v4f*)(dst + r * HID + 4) = o1;
  }
}

extern "C" void kernel_launch(void* const* d_in, const int* in_sizes, int n_in,
                              void* d_out, int out_size, void* d_ws, size_t ws_size,
                              hipStream_t stream) {
  (void)n_in; (void)d_ws; (void)ws_size; (void)out_size;
  const int*   ids = (const int*)d_in[0];
  const float* tok = (const float*)d_in[1];
  const float* pos = (const float*)d_in[2];
  const float* Wq  = (const float*)d_in[3];  const float* bq  = (const float*)d_in[4];
  const float* Wk  = (const float*)d_in[5];  const float* bk  = (const float*)d_in[6];
  const float* Wv  = (const float*)d_in[7];  const float* bv  = (const float*)d_in[8];
  const float* Wo  = (const float*)d_in[9];  const float* bo  = (const float*)d_in[10];
  const float* W1  = (const float*)d_in[11]; const float* b1  = (const float*)d_in[12];
  const float* W2  = (const float*)d_in[13]; const float* b2  = (const float*)d_in[14];
  const float* g1  = (const float*)d_in[15]; const float* be1 = (const float*)d_in[16];
  const float* g2  = (const float*)d_in[17]; const float* be2 = (const float*)d_in[18];
  float* out = (float*)d_out;

  int Btot = in_sizes[0] / SEQ;                       // 262144
  int batches_per_blk = WAVES_PER_BLK * BATCH_PER_WAVE; // 32
  int nblk = (Btot + batches_per_blk - 1) / batches_per_blk; // 8192
  tiny_transformer_fused<<<nblk, WAVES_PER_BLK * 32, 0, stream>>>(
      ids, tok, pos, Wq, bq, Wk, bk, Wv, bv, Wo, bo,
      W1, b1, W2, b2, g1, be1, g2, be2, out, Btot);
}